// NGCF_24481313587565
// MI455X (gfx1250) — compile-verified
//
#include <hip/hip_runtime.h>
#include <hip/hip_bf16.h>

#define N_USER_C 50000
#define N_ITEM_C 100000
#define N_TOT_C  150000
#define EMB_C    64

typedef __attribute__((ext_vector_type(2))) float v2f;
typedef __attribute__((ext_vector_type(4))) float v4f;
typedef __attribute__((ext_vector_type(8))) float v8f;

// ---------------------------------------------------------------------------
// ego = concat(user_emb, item_emb); total = ego
// ---------------------------------------------------------------------------
__global__ void ngcf_init(const float* __restrict__ ue, const float* __restrict__ ie,
                          float* __restrict__ ego, float* __restrict__ total) {
  int i = blockIdx.x * blockDim.x + threadIdx.x;
  if (i >= N_TOT_C * EMB_C) return;
  float v = (i < N_USER_C * EMB_C) ? ue[i] : ie[i - N_USER_C * EMB_C];
  ego[i]   = v;
  total[i] = v;
}

__global__ void ngcf_zero(float* __restrict__ p, int n) {
  int i = blockIdx.x * blockDim.x + threadIdx.x;
  if (i < n) p[i] = 0.f;
}

// ---------------------------------------------------------------------------
// side[row] += val * ego[col]  (scatter-add SpMM)
// One thread per (edge, 4-float chunk): b128 ego load + 4 hardware f32 atomics.
// Edge streams (rows/cols/vals) are non-temporal so ego/side/total stay
// resident in the 192MB L2; atomics land in L2.
// ---------------------------------------------------------------------------
__global__ void ngcf_spmm(const int* __restrict__ rows, const int* __restrict__ cols,
                          const float* __restrict__ vals, const float* __restrict__ ego,
                          float* __restrict__ side, int nnz) {
  long long gid = (long long)blockIdx.x * blockDim.x + threadIdx.x;
  int e = (int)(gid >> 4);
  if (e >= nnz) return;
  int c = (int)(gid & 15) * 4;
  int r = __builtin_nontemporal_load(rows + e);
  int s = __builtin_nontemporal_load(cols + e);
  float v = __builtin_nontemporal_load(vals + e);
  const v4f eg = *(const v4f*)(ego + s * EMB_C + c);
  float* dst = side + r * EMB_C + c;
  unsafeAtomicAdd(dst + 0, v * eg.x);
  unsafeAtomicAdd(dst + 1, v * eg.y);
  unsafeAtomicAdd(dst + 2, v * eg.z);
  unsafeAtomicAdd(dst + 3, v * eg.w);
}

// ---------------------------------------------------------------------------
// XOR-swizzle add across 16-lane groups (group-of-32 ds_swizzle, no LDS mem).
// offset = xor_mask<<10 | and_mask(0x1F)
// ---------------------------------------------------------------------------
__device__ __forceinline__ float swz_xor_add(float s, int offset_imm) {
  switch (offset_imm) {   // ds_swizzle offset must be an immediate
    case 1: s += __int_as_float(__builtin_amdgcn_ds_swizzle(__float_as_int(s), 0x041F)); break;
    case 2: s += __int_as_float(__builtin_amdgcn_ds_swizzle(__float_as_int(s), 0x081F)); break;
    case 4: s += __int_as_float(__builtin_amdgcn_ds_swizzle(__float_as_int(s), 0x101F)); break;
    case 8: s += __int_as_float(__builtin_amdgcn_ds_swizzle(__float_as_int(s), 0x201F)); break;
  }
  return s;
}

// ---------------------------------------------------------------------------
// Fused: ego_new = leaky_relu(side @ W + b, 0.2); total += ego_new / ||row||2
// One wave32 computes a 16x64 row-block via V_WMMA_F32_16X16X4_F32.
//   A (16x4 f32): lane L holds A[L%16][2*(L/16) + {0,1}]   (2 VGPRs)
//   B ( 4x16 f32): lane L holds B[2*(L/16) + {0,1}][L%16]  (2 VGPRs)
//   C/D (16x16 f32): vgpr i, lane L holds D[i + 8*(L/16)][L%16] (8 VGPRs)
// ---------------------------------------------------------------------------
__global__ void ngcf_gemm_act_norm(const float* __restrict__ side,
                                   const float* __restrict__ W,
                                   const float* __restrict__ bias,
                                   float* __restrict__ ego,
                                   float* __restrict__ total) {
  const int wave    = blockIdx.x * (blockDim.x >> 5) + (threadIdx.x >> 5);
  const int rowBase = wave * 16;
  if (rowBase >= N_TOT_C) return;              // wave-uniform: EXEC stays all-1s
  const int lane  = threadIdx.x & 31;
  const int lhalf = lane >> 4;                 // 0 | 1
  const int lmod  = lane & 15;

  v8f acc[4];
  acc[0] = (v8f){}; acc[1] = (v8f){}; acc[2] = (v8f){}; acc[3] = (v8f){};

  // K = 64 -> 16 steps of K=4
  for (int ks = 0; ks < 16; ++ks) {
    const int k0 = ks * 4 + lhalf * 2;
    const float* ap = side + (rowBase + lmod) * EMB_C + k0;
    v2f a = *(const v2f*)ap;                   // 8B-aligned (k0 even)
#pragma unroll
    for (int t = 0; t < 4; ++t) {
      const int n = t * 16 + lmod;
      v2f b;
      b.x = W[(k0    ) * EMB_C + n];
      b.y = W[(k0 + 1) * EMB_C + n];
      acc[t] = __builtin_amdgcn_wmma_f32_16x16x4_f32(
          /*neg_a=*/false, a, /*neg_b=*/false, b,
          /*c_mod=*/(short)0, acc[t], /*reuse_a=*/false, /*reuse_b=*/false);
    }
  }

  // bias + leaky_relu(0.2) + per-row sum of squares (partial, per lane)
  float sq[8];
#pragma unroll
  for (int i = 0; i < 8; ++i) sq[i] = 0.f;
#pragma unroll
  for (int t = 0; t < 4; ++t) {
    const float bv = bias[t * 16 + lmod];
#pragma unroll
    for (int i = 0; i < 8; ++i) {
      float x = acc[t][i] + bv;
      x = (x > 0.f) ? x : 0.2f * x;
      acc[t][i] = x;
      sq[i] += x * x;
    }
  }

  // reduce sum-of-squares across the 16 lanes holding each row (XOR swizzles)
#pragma unroll
  for (int i = 0; i < 8; ++i) {
    float s = sq[i];
    s = swz_xor_add(s, 1);
    s = swz_xor_add(s, 2);
    s = swz_xor_add(s, 4);
    s = swz_xor_add(s, 8);
    sq[i] = 1.f / fmaxf(sqrtf(s), 1e-12f);     // inverse L2 norm of row i+8*lhalf
  }

  // write ego (raw activation) and total += normalized activation
#pragma unroll
  for (int i = 0; i < 8; ++i) {
    const int row = rowBase + i + 8 * lhalf;
    const float inv = sq[i];
#pragma unroll
    for (int t = 0; t < 4; ++t) {
      const int col = t * 16 + lmod;
      const float x = acc[t][i];
      ego[row * EMB_C + col] = x;
      total[row * EMB_C + col] += x * inv;
    }
  }
}

// ---------------------------------------------------------------------------
// out[b] = total[users[b]]
// ---------------------------------------------------------------------------
__global__ void ngcf_gather(const int* __restrict__ users, const float* __restrict__ total,
                            float* __restrict__ out, int batch) {
  int i = blockIdx.x * blockDim.x + threadIdx.x;
  if (i >= batch * EMB_C) return;
  int b = i >> 6;
  int c = i & 63;
  out[i] = total[users[b] * EMB_C + c];
}

// ---------------------------------------------------------------------------
extern "C" void kernel_launch(void* const* d_in, const int* in_sizes, int n_in,
                              void* d_out, int out_size, void* d_ws, size_t ws_size,
                              hipStream_t stream) {
  const int*   users    = (const int*)d_in[0];
  const int*   rows     = (const int*)d_in[1];
  const int*   cols     = (const int*)d_in[2];
  const float* vals     = (const float*)d_in[3];
  const float* user_emb = (const float*)d_in[4];
  const float* item_emb = (const float*)d_in[5];
  const float* Ws[3] = {(const float*)d_in[6], (const float*)d_in[8], (const float*)d_in[10]};
  const float* bs[3] = {(const float*)d_in[7], (const float*)d_in[9], (const float*)d_in[11]};
  float* out = (float*)d_out;

  const int nnz   = in_sizes[1];
  const int batch = in_sizes[0];
  const size_t mat = (size_t)N_TOT_C * EMB_C;   // 9.6M floats = 38.4 MB

  float* ego   = (float*)d_ws;
  float* side  = ego  + mat;
  float* total = side + mat;

  const int elems = N_TOT_C * EMB_C;
  ngcf_init<<<(elems + 255) / 256, 256, 0, stream>>>(user_emb, item_emb, ego, total);

  const long long spmm_threads = (long long)nnz * 16;   // 4 floats / thread
  const int spmm_blocks = (int)((spmm_threads + 255) / 256);
  const int waves = (N_TOT_C + 15) / 16;                // 9375
  const int gemm_blocks = (waves + 3) / 4;              // 4 waves / 128-thread block

  for (int l = 0; l < 3; ++l) {
    ngcf_zero<<<(elems + 255) / 256, 256, 0, stream>>>(side, elems);
    ngcf_spmm<<<spmm_blocks, 256, 0, stream>>>(rows, cols, vals, ego, side, nnz);
    ngcf_gemm_act_norm<<<gemm_blocks, 128, 0, stream>>>(side, Ws[l], bs[l], ego, total);
  }

  ngcf_gather<<<(batch * EMB_C + 255) / 256, 256, 0, stream>>>(users, total, out, batch);
}